// SoPBC_54451595378998
// MI455X (gfx1250) — compile-verified
//
#include <hip/hip_runtime.h>
#include <hip/hip_bf16.h>

// Problem constants (from reference)
#define B_    4
#define W_    4096
#define N_    2
#define S_    449
#define LH    25
#define WOUT  4046           // W - 2*LH
#define WAVE_W 8             // output positions per wave (A rows = 8 w x 2 pol)
#define WG_W  64             // 8 waves * 8 outputs
#define BLOCKS_PER_BATCH 64  // ceil(4046/64)

typedef __attribute__((ext_vector_type(2))) float v2f;
typedef __attribute__((ext_vector_type(8))) float v8f;

__global__ __launch_bounds__(256)
void SoPBC_kernel(const float* __restrict__ E_real,
                  const float* __restrict__ E_imag,
                  const float* __restrict__ C_real,
                  const float* __restrict__ C_imag,
                  const float* __restrict__ task_info,
                  const int*   __restrict__ m_idx,
                  const int*   __restrict__ n_idx,
                  float*       __restrict__ out)
{
    // E window: position t holds ring index (o0 + LH - 64 + t) & 4095,
    // packed as (re0, im0, re1, im1) per position.
    __shared__ float4 ewin[256];
    __shared__ float  c_re[S_ + 3];
    __shared__ float  c_im[S_ + 3];
    __shared__ int    m_l [S_ + 3];
    __shared__ int    n_l [S_ + 3];

    const int tid = threadIdx.x;
    const int b   = blockIdx.x / BLOCKS_PER_BATCH;
    const int o0  = (blockIdx.x % BLOCKS_PER_BATCH) * WG_W;

    // ---- stage E window (ring wrap handled here) ----
    {
        int kpos = (o0 + LH - 64 + tid) & (W_ - 1);
        const float2* er = (const float2*)(E_real + ((size_t)(b * W_ + kpos)) * N_);
        const float2* ei = (const float2*)(E_imag + ((size_t)(b * W_ + kpos)) * N_);
        float2 r = *er;
        float2 i = *ei;
        ewin[tid] = make_float4(r.x, i.x, r.y, i.y);
    }
    // ---- stage shift indices & complex coefficients ----
    for (int t = tid; t < S_; t += 256) {
        c_re[t] = C_real[t];
        c_im[t] = C_imag[t];
        m_l[t]  = m_idx[t];
        n_l[t]  = n_idx[t];
    }
    __syncthreads();

    const int wave  = tid >> 5;
    const int lane  = tid & 31;
    const int row   = lane & 15;   // A-matrix row M == B/D column N for this lane
    const int khalf = lane >> 4;   // 0: K=0,1 (shift 2c) ; 1: K=2,3 (shift 2c+1)
    const int wl    = row >> 1;    // local output position within wave tile (0..7)
    const int nn    = row & 1;     // polarization
    const int klocal = 64 + wave * WAVE_W + wl;  // LDS slot of this row's ring pos

    v8f acc = {};                  // D/C fragment, accumulates over all chunks
    const int nchunk = (S_ + 1) / 2;   // 225

    for (int c = 0; c < nchunk; ++c) {
        int   sraw = 2 * c + khalf;
        float wv   = (sraw < S_) ? 1.0f : 0.0f;   // zero-weight the odd tail
        int   s    = (sraw < S_) ? sraw : (S_ - 1);
        int   ms   = m_l[s];
        int   ns   = n_l[s];
        float Cr   = c_re[s] * wv;
        float Ci   = c_im[s] * wv;

        float4 En  = ewin[klocal - ns];
        float4 Emn = ewin[klocal - ms - ns];
        float4 Em  = ewin[klocal - ms];

        // A_n = En_n * conj(Emn_n); Ssum = A_0 + A_1 (shared by both pols)
        float a0r = En.x * Emn.x + En.y * Emn.y;
        float a0i = En.y * Emn.x - En.x * Emn.y;
        float a1r = En.z * Emn.z + En.w * Emn.w;
        float a1i = En.w * Emn.z - En.z * Emn.w;
        float sr  = a0r + a1r;
        float si  = a0i + a1i;
        float emr = nn ? Em.z : Em.x;
        float emi = nn ? Em.w : Em.y;

        v2f a, bb;
        a.x = sr * emr - si * emi;   // F_re  -> K = 2*khalf
        a.y = sr * emi + si * emr;   // F_im  -> K = 2*khalf + 1
        // B columns 0/1 = complex-multiply matrix [[Cr, Ci], [-Ci, Cr]]
        bb.x = (row == 0) ?  Cr : ((row == 1) ? Ci : 0.0f);
        bb.y = (row == 0) ? -Ci : ((row == 1) ? Cr : 0.0f);

        // D = A(16x4) x B(4x16) + C : corr_{re,im} accumulate in cols 0/1
        acc = __builtin_amdgcn_wmma_f32_16x16x4_f32(
                  false, a, false, bb, (short)0, acc, false, false);
    }

    // ---- epilogue: out = P * corr + E_center ----
    float P = expf(task_info[b * 4] * 0.23025850929940457f) * 0.5f; // 10^(x/10)/N

    if (row < 2) {   // only columns 0 (re) and 1 (im) of D are meaningful
        #pragma unroll
        for (int r = 0; r < 8; ++r) {
            int M   = 8 * khalf + r;        // D rows for this lane half
            int wl2 = M >> 1;
            int n2  = M & 1;
            int o   = o0 + wave * WAVE_W + wl2;
            if (o < WOUT) {
                float4 ec = ewin[64 + wave * WAVE_W + wl2];
                float  e  = (row == 0) ? (n2 ? ec.z : ec.x)
                                       : (n2 ? ec.w : ec.y);
                out[(((size_t)b * WOUT + o) * N_ + n2) * 2 + row] = P * acc[r] + e;
            }
        }
    }
}

extern "C" void kernel_launch(void* const* d_in, const int* in_sizes, int n_in,
                              void* d_out, int out_size, void* d_ws, size_t ws_size,
                              hipStream_t stream) {
    const float* E_real    = (const float*)d_in[0];
    const float* E_imag    = (const float*)d_in[1];
    const float* C_real    = (const float*)d_in[2];
    const float* C_imag    = (const float*)d_in[3];
    const float* task_info = (const float*)d_in[4];
    const int*   m_idx     = (const int*)d_in[5];
    const int*   n_idx     = (const int*)d_in[6];
    float*       outp      = (float*)d_out;

    dim3 grid(B_ * BLOCKS_PER_BATCH);   // 256 blocks
    dim3 block(256);                    // 8 waves of 32
    SoPBC_kernel<<<grid, block, 0, stream>>>(E_real, E_imag, C_real, C_imag,
                                             task_info, m_idx, n_idx, outp);
}